// AttentionModel_47966194761769
// MI455X (gfx1250) — compile-verified
//
#include <hip/hip_runtime.h>

// ---------------------------------------------------------------------------
// Problem: B=32, S=4096, H=1024, F=1024 (all fp32)
//   query    = feature @ W_q + b_q                      [B,H]
//   scores   = einsum('bh,bsh->bs', query, hidden)      [B,S]
//   attn     = softmax(scores, axis=S)
//   attended = einsum('bs,bsh->bh', attn, hidden)       [B,H]
//
// Strategy: (1) WMMA f32 16x16x4 for the query GEMM,
//           (2) online-softmax fused pass reading hidden ONCE (register tiles),
//           (3) small log-sum-exp merge reduce.
// ---------------------------------------------------------------------------

#define BATCH 32
#define SEQ   4096
#define HID   1024
#define FEAT  1024

#define WAVES_PER_BATCH 64          // split-K groups over S
#define ROWS_PER_WAVE   (SEQ / WAVES_PER_BATCH)   // 64

typedef float v2f __attribute__((ext_vector_type(2)));
typedef float v8f __attribute__((ext_vector_type(8)));

// ---------------------------------------------------------------------------
// Kernel 1: query[32,1024] = feature[32,1024] @ W_q[1024,1024] + b_q
// One wave (32 threads) per 16x16 output tile; K loop in steps of 4 using
// V_WMMA_F32_16X16X4_F32.
//
// A-frag 16x4 f32 layout: lanes 0-15 hold M=lane, K={k0,k0+1}; lanes 16-31
// hold M=lane-16, K={k0+2,k0+3}. B-frag mirrors with N across lanes.
// C/D 16x16 f32: VGPR i -> M = i (lanes 0-15) / i+8 (lanes 16-31), N = lane%16.
// ---------------------------------------------------------------------------
__global__ __launch_bounds__(32) void query_wmma_kernel(
    const float* __restrict__ F,   // [32,1024]
    const float* __restrict__ W,   // [1024,1024]
    const float* __restrict__ bq,  // [1024]
    float* __restrict__ q)         // [32,1024]
{
    const int tile = blockIdx.x;       // 0..127  (2 row tiles x 64 col tiles)
    const int tm   = tile >> 6;        // 0..1
    const int tn   = tile & 63;        // 0..63
    const int lane = threadIdx.x;      // 0..31
    const int half = lane >> 4;        // 0: K lo pair, 1: K hi pair
    const int l16  = lane & 15;
    const int m    = tm * 16 + l16;    // A row handled by this lane
    const int n    = tn * 16 + l16;    // B col handled by this lane

    v8f c = {};
    for (int k0 = 0; k0 < FEAT; k0 += 4) {
        const int ka = k0 + half * 2;
        v2f a;
        a.x = F[m * FEAT + ka + 0];
        a.y = F[m * FEAT + ka + 1];
        v2f bf;
        bf.x = W[(ka + 0) * HID + n];
        bf.y = W[(ka + 1) * HID + n];
        c = __builtin_amdgcn_wmma_f32_16x16x4_f32(
                /*neg_a=*/false, a, /*neg_b=*/false, bf,
                /*c_mod=*/(short)0, c, /*reuse_a=*/false, /*reuse_b=*/false);
    }

    const float bias = bq[n];
#pragma unroll
    for (int i = 0; i < 8; ++i) {
        const int row = tm * 16 + i + half * 8;
        q[row * HID + tn * 16 + l16] = c[i] + bias;
    }
}

// ---------------------------------------------------------------------------
// Kernel 2: fused online-softmax partial pass. Each wave owns 64 rows of S for
// one batch. Row is held in registers (8 x float4 per lane = 128B/lane),
// score = wave-reduced dot(q,row), then folded into a register accumulator
// with running-max rescaling. hidden is read from HBM exactly once.
// Outputs per wave: acc[1024], m, l.
// ---------------------------------------------------------------------------
__global__ __launch_bounds__(256) void flash_partial_kernel(
    const float* __restrict__ hid,   // [32,4096,1024]
    const float* __restrict__ q,     // [32,1024]
    float* __restrict__ pacc,        // [32*64, 1024]
    float* __restrict__ pml)         // [32*64, 2] (m,l)
{
    const int b     = blockIdx.y;                 // batch
    const int wave  = threadIdx.x >> 5;           // 0..7
    const int lane  = threadIdx.x & 31;
    const int wslot = blockIdx.x * 8 + wave;      // 0..63 within batch

    // query fragments: lane covers h = j*128 + lane*4 .. +3
    const float* qb = q + b * HID;
    float4 qv[8];
#pragma unroll
    for (int j = 0; j < 8; ++j)
        qv[j] = *(const float4*)(qb + j * 128 + lane * 4);

    float m = -3.0e38f;
    float l = 0.0f;
    float4 acc[8];
#pragma unroll
    for (int j = 0; j < 8; ++j)
        acc[j] = make_float4(0.f, 0.f, 0.f, 0.f);

    const float* base = hid + ((size_t)b * SEQ + (size_t)wslot * ROWS_PER_WAVE) * HID;

    for (int s = 0; s < ROWS_PER_WAVE; ++s) {
        const float* row = base + (size_t)s * HID;
        float4 rv[8];
#pragma unroll
        for (int j = 0; j < 8; ++j)
            rv[j] = *(const float4*)(row + j * 128 + lane * 4);

        // per-lane partial dot
        float p = 0.f;
#pragma unroll
        for (int j = 0; j < 8; ++j) {
            p += qv[j].x * rv[j].x;
            p += qv[j].y * rv[j].y;
            p += qv[j].z * rv[j].z;
            p += qv[j].w * rv[j].w;
        }
        // wave32 all-reduce (butterfly)
#pragma unroll
        for (int off = 16; off > 0; off >>= 1)
            p += __shfl_xor(p, off, 32);

        // online softmax update
        const float mn = fmaxf(m, p);
        const float so = __expf(m - mn);   // rescale old state (0 on first iter)
        const float e  = __expf(p - mn);
        l = l * so + e;
#pragma unroll
        for (int j = 0; j < 8; ++j) {
            acc[j].x = acc[j].x * so + e * rv[j].x;
            acc[j].y = acc[j].y * so + e * rv[j].y;
            acc[j].z = acc[j].z * so + e * rv[j].z;
            acc[j].w = acc[j].w * so + e * rv[j].w;
        }
        m = mn;
    }

    const int pidx = b * WAVES_PER_BATCH + wslot;
    float* dst = pacc + (size_t)pidx * HID;
#pragma unroll
    for (int j = 0; j < 8; ++j)
        *(float4*)(dst + j * 128 + lane * 4) = acc[j];
    if (lane == 0) {
        pml[2 * pidx + 0] = m;
        pml[2 * pidx + 1] = l;
    }
}

// ---------------------------------------------------------------------------
// Kernel 3: merge 64 partials per batch:
//   M = max_w m_w ; wgt_w = exp(m_w - M) ; L = sum_w wgt_w * l_w
//   out[b,h] = (sum_w wgt_w * acc_w[h]) / L
// ---------------------------------------------------------------------------
__global__ __launch_bounds__(256) void flash_reduce_kernel(
    const float* __restrict__ pacc,  // [32*64, 1024]
    const float* __restrict__ pml,   // [32*64, 2]
    float* __restrict__ out)         // [32,1024]
{
    const int b = blockIdx.x;
    const int t = threadIdx.x;

    __shared__ float wgt[WAVES_PER_BATCH];
    __shared__ float invL;

    if (t == 0) {
        float M = -3.0e38f;
        for (int w = 0; w < WAVES_PER_BATCH; ++w)
            M = fmaxf(M, pml[2 * (b * WAVES_PER_BATCH + w)]);
        float L = 0.f;
        for (int w = 0; w < WAVES_PER_BATCH; ++w) {
            const float e = __expf(pml[2 * (b * WAVES_PER_BATCH + w)] - M);
            wgt[w] = e;
            L += e * pml[2 * (b * WAVES_PER_BATCH + w) + 1];
        }
        invL = 1.0f / L;
    }
    __syncthreads();

    for (int h = t; h < HID; h += 256) {
        float sum = 0.f;
#pragma unroll 4
        for (int w = 0; w < WAVES_PER_BATCH; ++w)
            sum += wgt[w] * pacc[(size_t)(b * WAVES_PER_BATCH + w) * HID + h];
        out[b * HID + h] = sum * invL;
    }
}

// ---------------------------------------------------------------------------
extern "C" void kernel_launch(void* const* d_in, const int* in_sizes, int n_in,
                              void* d_out, int out_size, void* d_ws, size_t ws_size,
                              hipStream_t stream) {
    const float* F   = (const float*)d_in[0];   // feature_vector [32,1024]
    const float* hid = (const float*)d_in[1];   // last_hidden_state [32,4096,1024]
    const float* W   = (const float*)d_in[2];   // W_q [1024,1024]
    const float* bq  = (const float*)d_in[3];   // b_q [1024]
    float* out = (float*)d_out;                 // [32,1024]

    // workspace layout (floats): query | partial acc | partial (m,l)
    float* q    = (float*)d_ws;                                   // 32*1024
    float* pacc = q + BATCH * HID;                                // 2048*1024
    float* pml  = pacc + (size_t)BATCH * WAVES_PER_BATCH * HID;   // 2048*2
    (void)in_sizes; (void)n_in; (void)out_size; (void)ws_size;

    // 1) query = feature @ W_q + b_q   (WMMA f32)
    query_wmma_kernel<<<dim3((BATCH / 16) * (HID / 16)), 32, 0, stream>>>(F, W, bq, q);

    // 2) fused scores/softmax/weighted-sum partials (single read of hidden)
    flash_partial_kernel<<<dim3(WAVES_PER_BATCH / 8, BATCH), 256, 0, stream>>>(hid, q, pacc, pml);

    // 3) merge partials
    flash_reduce_kernel<<<dim3(BATCH), 256, 0, stream>>>(pacc, pml, out);
}